// GraphAttentionSparse_11433202942857
// MI455X (gfx1250) — compile-verified
//
#include <hip/hip_runtime.h>
#include <math.h>

#define B_   4
#define N_   4096
#define C_   256
#define H_   8
#define D_   32
#define HD_  256
#define KNN_ 16
#define JCH  256   // j-chunk width in knn kernel (4 waves x 64 cols)
#define NB_  4     // 16-col subtiles per wave (register blocking)

typedef __attribute__((ext_vector_type(2))) float v2f;
typedef __attribute__((ext_vector_type(8))) float v8f;

// ---------------------------------------------------------------------------
// Row squared norms: one wave per row.
// ---------------------------------------------------------------------------
__global__ __launch_bounds__(256) void sq_kernel(const float* __restrict__ x,
                                                 float* __restrict__ sq) {
  int wid  = (int)((blockIdx.x * 256 + threadIdx.x) >> 5);
  int lane = (int)(threadIdx.x & 31);
  if (wid >= B_ * N_) return;
  const float* row = x + (size_t)wid * C_;
  float s = 0.f;
#pragma unroll
  for (int k = 0; k < C_ / 32; ++k) {
    float v = row[lane + k * 32];
    s = fmaf(v, v, s);
  }
#pragma unroll
  for (int off = 16; off; off >>= 1) s += __shfl_xor(s, off, 32);
  if (lane == 0) sq[wid] = s;
}

// ---------------------------------------------------------------------------
// Q/K/V projections: one wave computes a 16x64 output tile with
// V_WMMA_F32_16X16X4_F32, register-blocked 4x in N so each A fetch feeds
// 4 WMMAs. blockIdx.z selects Q/K/V.
// A (16x4 f32): lanes 0-15 hold K=0,1; lanes 16-31 hold K=2,3 (row M=lane&15)
// B (4x16 f32): lanes 0-15 hold K=0,1; lanes 16-31 hold K=2,3 (col N=lane&15)
// C/D (16x16):  VGPR r -> M = r + 8*(lane>=16), N = lane&15
// ---------------------------------------------------------------------------
__global__ __launch_bounds__(32) void proj_kernel(
    const float* __restrict__ x,
    const float* __restrict__ Wq, const float* __restrict__ bq,
    const float* __restrict__ Wk, const float* __restrict__ bk,
    const float* __restrict__ Wv, const float* __restrict__ bv,
    float* __restrict__ Q, float* __restrict__ Kn, float* __restrict__ Vn) {
  const float* W; const float* bias; float* out;
  if (blockIdx.z == 0)      { W = Wq; bias = bq; out = Q;  }
  else if (blockIdx.z == 1) { W = Wk; bias = bk; out = Kn; }
  else                      { W = Wv; bias = bv; out = Vn; }

  int rowBase = (int)blockIdx.x * 16;
  int colBase = (int)blockIdx.y * 64;
  int lane    = (int)(threadIdx.x & 31);
  int mn      = lane & 15;          // row for A, col-within-subtile for B/D
  int koff    = (lane >> 4) * 2;    // lane-half K offset

  const float* aPtr = x + (size_t)(rowBase + mn) * C_ + koff;
  const float* bPtr = W + (size_t)koff * HD_ + colBase + mn;

  v8f acc[NB_];
#pragma unroll
  for (int t = 0; t < NB_; ++t) acc[t] = (v8f){};

  for (int k = 0; k < C_; k += 4) {
    v2f a = *(const v2f*)(aPtr + k);
    __builtin_prefetch(aPtr + k + 16, 0, 1);
#pragma unroll
    for (int t = 0; t < NB_; ++t) {
      v2f b;
      b[0] = bPtr[(size_t)k * HD_ + t * 16];
      b[1] = bPtr[(size_t)(k + 1) * HD_ + t * 16];
      acc[t] = __builtin_amdgcn_wmma_f32_16x16x4_f32(false, a, false, b,
                                                     (short)0, acc[t],
                                                     false, false);
    }
  }

  int mrow = (lane >> 4) * 8;
#pragma unroll
  for (int t = 0; t < NB_; ++t) {
    float bb = bias[colBase + t * 16 + mn];
#pragma unroll
    for (int r = 0; r < 8; ++r) {
      out[(size_t)(rowBase + mrow + r) * HD_ + colBase + t * 16 + mn] =
          acc[t][r] + bb;
    }
  }
}

// ---------------------------------------------------------------------------
// Fused KNN: distance GEMM (WMMA f32, 16x64 per wave, A from LDS reused
// across 4 subtiles) + running top-16; the NxN distance matrix is never
// materialized. Workgroup = 4 waves = one 16-row target tile, 256-col chunks.
// ---------------------------------------------------------------------------
__global__ __launch_bounds__(128) void knn_kernel(const float* __restrict__ x,
                                                  const float* __restrict__ sq,
                                                  int* __restrict__ idxOut) {
  __shared__ float sXi[16 * C_];      // staged target-row tile (16 KB)
  __shared__ float sSqi[16];
  __shared__ float sD2[16 * JCH];     // distance tile for current chunk (16 KB)
  __shared__ float sCd[16 * 128];     // per-row candidates (8 threads x 16)
  __shared__ int   sCi[16 * 128];

  int tid   = (int)threadIdx.x;
  int wave  = tid >> 5;
  int lane  = tid & 31;
  int tile  = (int)blockIdx.x;                 // 0 .. B*N/16-1
  int b     = tile / (N_ / 16);
  int iBase = (tile % (N_ / 16)) * 16;
  const float* xb  = x  + (size_t)b * N_ * C_;
  const float* sqb = sq + (size_t)b * N_;

  // stage x_i tile into LDS (16 rows x 256 floats)
  for (int t = tid; t < 16 * C_ / 4; t += 128)
    ((float4*)sXi)[t] = ((const float4*)(xb + (size_t)iBase * C_))[t];
  if (tid < 16) sSqi[tid] = sqb[iBase + tid];
  __syncthreads();

  float bestD[KNN_]; int bestI[KNN_];
#pragma unroll
  for (int t = 0; t < KNN_; ++t) { bestD[t] = INFINITY; bestI[t] = -1; }

  int mn     = lane & 15;
  int koff   = (lane >> 4) * 2;
  int mrow   = (lane >> 4) * 8;
  int myRow  = tid >> 3;   // 0..15
  int mySlot = tid & 7;    // 0..7
  const float* aLds = &sXi[(size_t)mn * C_ + koff];

  for (int jBase = 0; jBase < N_; jBase += JCH) {
    int jw = jBase + wave * 64;                // this wave's 64 columns
    const float* bPtr = xb + (size_t)(jw + mn) * C_ + koff;

    v8f acc[NB_];
#pragma unroll
    for (int t = 0; t < NB_; ++t) acc[t] = (v8f){};

    for (int k = 0; k < C_; k += 4) {
      v2f a = *(const v2f*)(aLds + k);               // ds_read_b64, reused 4x
#pragma unroll
      for (int t = 0; t < NB_; ++t) {
        v2f bb = *(const v2f*)(bPtr + (size_t)t * 16 * C_ + k);
        acc[t] = __builtin_amdgcn_wmma_f32_16x16x4_f32(false, a, false, bb,
                                                       (short)0, acc[t],
                                                       false, false);
      }
    }

    // d2 = |xi|^2 + |xj|^2 - 2*dot; exclude self; write tile to LDS
#pragma unroll
    for (int t = 0; t < NB_; ++t) {
      int colInChunk = wave * 64 + t * 16 + mn;
      float sqj = sqb[jBase + colInChunk];
#pragma unroll
      for (int r = 0; r < 8; ++r) {
        int mi = mrow + r;
        float d2 = sSqi[mi] + sqj - 2.0f * acc[t][r];
        if (iBase + mi == jBase + colInChunk) d2 = INFINITY;
        sD2[mi * JCH + colInChunk] = d2;
      }
    }
    __syncthreads();

    // running top-16: thread (myRow, mySlot) scans 32 candidates of its row
#pragma unroll 4
    for (int u = 0; u < JCH / 8; ++u) {
      int jj = mySlot * (JCH / 8) + u;
      float d = sD2[myRow * JCH + jj];
      if (d < bestD[KNN_ - 1]) {
        float v = d; int vi = jBase + jj;
#pragma unroll
        for (int t = 0; t < KNN_; ++t) {       // sorted insert (branch-free)
          bool sw = v < bestD[t];
          float tv = bestD[t]; int ti = bestI[t];
          bestD[t] = sw ? v : tv;  bestI[t] = sw ? vi : ti;
          v = sw ? tv : v;         vi = sw ? ti : vi;
        }
      }
    }
    __syncthreads();
  }

  // merge 8 per-thread lists per row -> global top-16 of that row
#pragma unroll
  for (int t = 0; t < KNN_; ++t) {
    sCd[myRow * 128 + mySlot * KNN_ + t] = bestD[t];
    sCi[myRow * 128 + mySlot * KNN_ + t] = bestI[t];
  }
  __syncthreads();

  if (tid < 16) {
    int row = tid;
    int* op = idxOut + (size_t)(b * N_ + iBase + row) * KNN_;
    for (int s = 0; s < KNN_; ++s) {
      float mBest = INFINITY; int mPos = 0;
      for (int c = 0; c < 128; ++c) {
        float d = sCd[row * 128 + c];
        if (d < mBest) { mBest = d; mPos = c; }
      }
      op[s] = sCi[row * 128 + mPos];
      sCd[row * 128 + mPos] = INFINITY;
    }
  }
}

// ---------------------------------------------------------------------------
// Sparse attention over the 16 neighbors: one wave per (node, head),
// lanes = the 32 head dims.
// ---------------------------------------------------------------------------
__global__ __launch_bounds__(128) void attn_kernel(
    const float* __restrict__ Q, const float* __restrict__ Kn,
    const float* __restrict__ Vn, const int* __restrict__ idx,
    float* __restrict__ out) {
  int gwid = (int)((blockIdx.x * 128 + threadIdx.x) >> 5);
  int lane = (int)(threadIdx.x & 31);
  int h    = gwid & (H_ - 1);
  int node = gwid >> 3;                 // / H_
  if (node >= B_ * N_) return;
  int b = node / N_;

  const int* nb = idx + (size_t)node * KNN_;
  size_t baseQ  = (size_t)node * HD_ + (size_t)h * D_;
  float qv = Q[baseQ + lane];
  const float scale = 0.17677669529663687f;   // 1/sqrt(32)

  float sc[KNN_];
#pragma unroll
  for (int j = 0; j < KNN_; ++j) {
    int nj = nb[j];
    float kv = Kn[(size_t)(b * N_ + nj) * HD_ + (size_t)h * D_ + lane];
    float p = qv * kv;
#pragma unroll
    for (int off = 16; off; off >>= 1) p += __shfl_xor(p, off, 32);
    sc[j] = p * scale;
  }

  float mx = -INFINITY;
#pragma unroll
  for (int j = 0; j < KNN_; ++j) mx = fmaxf(mx, sc[j]);
  float ssum = 0.f;
#pragma unroll
  for (int j = 0; j < KNN_; ++j) { sc[j] = __expf(sc[j] - mx); ssum += sc[j]; }
  float inv = 1.0f / ssum;

  float acc = 0.f;
#pragma unroll
  for (int j = 0; j < KNN_; ++j) {
    int nj = nb[j];
    acc = fmaf(sc[j] * inv,
               Vn[(size_t)(b * N_ + nj) * HD_ + (size_t)h * D_ + lane], acc);
  }
  out[baseQ + lane] = acc;
}

// ---------------------------------------------------------------------------
extern "C" void kernel_launch(void* const* d_in, const int* in_sizes, int n_in,
                              void* d_out, int out_size, void* d_ws, size_t ws_size,
                              hipStream_t stream) {
  const float* x  = (const float*)d_in[0];
  const float* Wq = (const float*)d_in[1];
  const float* bq = (const float*)d_in[2];
  const float* Wk = (const float*)d_in[3];
  const float* bk = (const float*)d_in[4];
  const float* Wv = (const float*)d_in[5];
  const float* bv = (const float*)d_in[6];
  float* out = (float*)d_out;

  char* ws = (char*)d_ws;
  float* sq = (float*)ws; ws += (size_t)B_ * N_ * sizeof(float);
  float* Q  = (float*)ws; ws += (size_t)B_ * N_ * HD_ * sizeof(float);
  float* Kn = (float*)ws; ws += (size_t)B_ * N_ * HD_ * sizeof(float);
  float* Vn = (float*)ws; ws += (size_t)B_ * N_ * HD_ * sizeof(float);
  int*  idx = (int*)ws;   ws += (size_t)B_ * N_ * KNN_ * sizeof(int);

  sq_kernel<<<B_ * N_ / 8, 256, 0, stream>>>(x, sq);

  dim3 pgrid(B_ * N_ / 16, HD_ / 64, 3);
  proj_kernel<<<pgrid, 32, 0, stream>>>(x, Wq, bq, Wk, bk, Wv, bv, Q, Kn, Vn);

  knn_kernel<<<B_ * N_ / 16, 128, 0, stream>>>(x, sq, idx);

  attn_kernel<<<(B_ * N_ * H_) / 4, 128, 0, stream>>>(Q, Kn, Vn, idx, out);
}